// SigNet__I_Block_42571715838544
// MI455X (gfx1250) — compile-verified
//
#include <hip/hip_runtime.h>
#include <stdint.h>

// ---------------------------------------------------------------------------
// Types for f32 WMMA (V_WMMA_F32_16X16X4_F32): A=16x4 (2 VGPR), B=4x16 (2 VGPR),
// C/D = 16x16 f32 (8 VGPR).
// ---------------------------------------------------------------------------
typedef float v2f __attribute__((ext_vector_type(2)));
typedef float v8f __attribute__((ext_vector_type(8)));

// ---------------------------------------------------------------------------
// gfx1250 async global->LDS copies (ASYNCcnt-tracked). LDS offset = low 32
// bits of the flat address of a __shared__ object.
// ---------------------------------------------------------------------------
__device__ __forceinline__ void async_ld_f32(float* lds_dst, const float* gsrc) {
  uint32_t loff = (uint32_t)(uintptr_t)lds_dst;
  asm volatile("global_load_async_to_lds_b32 %0, %1, off"
               :: "v"(loff), "v"(gsrc) : "memory");
}
__device__ __forceinline__ void async_ld_f32x4(float* lds_dst, const float* gsrc) {
  uint32_t loff = (uint32_t)(uintptr_t)lds_dst;
  asm volatile("global_load_async_to_lds_b128 %0, %1, off"
               :: "v"(loff), "v"(gsrc) : "memory");
}
__device__ __forceinline__ void wait_async0() {
  asm volatile("s_wait_asynccnt 0x0" ::: "memory");
}
__device__ __forceinline__ void wait_async3() {   // allow 3 in-flight (next chunk)
  asm volatile("s_wait_asynccnt 0x3" ::: "memory");
}

// ---------------------------------------------------------------------------
// Generic GEMM with f32 WMMA:  Y[M,N] = act( X[M,K] @ W[N,K]^T + bias[N] )
// Block = 256 threads = 8 waves arranged 2(M)x4(N): block tile 32x64, each
// wave owns one 16x16 tile; 8 v_wmma_f32_16x16x4_f32 per 32-wide K chunk.
// Full tiles with K%32==0 use an async-b128, double-buffered LDS pipeline
// (issue next chunk, s_wait_asynccnt 3, compute current). Edge tiles use a
// guarded scalar path.
// ---------------------------------------------------------------------------
#define GEMM_KC 32
#define GEMM_LDP 36   // LDS row stride (floats): 16B-aligned rows, conflict-free

__global__ __launch_bounds__(256)
void wmma_gemm_bias_act(const float* __restrict__ X, const float* __restrict__ W,
                        const float* __restrict__ bias, float* __restrict__ Y,
                        int M, int N, int K, int relu) {
  __shared__ float As[2][32][GEMM_LDP];
  __shared__ float Ws[2][64][GEMM_LDP];
  const int tid  = threadIdx.x;
  const int lane = tid & 31;
  const int w    = tid >> 5;
  const int wm   = w >> 2;        // 0..1
  const int wn   = w & 3;         // 0..3
  const int m0   = blockIdx.x * 32;
  const int n0   = blockIdx.y * 64;

  v8f acc = {};
  const int row = lane & 15;
  const int kp  = (lane >> 4) << 1;   // lanes 0-15: K+0/K+1, lanes 16-31: K+2/K+3

  const bool fast = ((K & (GEMM_KC - 1)) == 0) && (m0 + 32 <= M) && (n0 + 64 <= N);

  if (fast) {
    const int lr = tid >> 3;            // 0..31: row within tile
    const int lc = (tid & 7) << 2;      // 0,4,...,28: 4-float column group
    const float* Xa  = X + (size_t)(m0 + lr) * K + lc;
    const float* Wb0 = W + (size_t)(n0 + lr) * K + lc;
    const float* Wb1 = W + (size_t)(n0 + 32 + lr) * K + lc;

    // prologue: stage chunk 0 into buffer 0
    async_ld_f32x4(&As[0][lr][lc], Xa);
    async_ld_f32x4(&Ws[0][lr][lc], Wb0);
    async_ld_f32x4(&Ws[0][32 + lr][lc], Wb1);

    int buf = 0;
    for (int k0 = 0; k0 < K; k0 += GEMM_KC) {
      const bool has_next = (k0 + GEMM_KC) < K;
      if (has_next) {
        const int nb = buf ^ 1, kn = k0 + GEMM_KC;
        async_ld_f32x4(&As[nb][lr][lc], Xa + kn);
        async_ld_f32x4(&Ws[nb][lr][lc], Wb0 + kn);
        async_ld_f32x4(&Ws[nb][32 + lr][lc], Wb1 + kn);
        wait_async3();                   // current chunk complete, next in flight
      } else {
        wait_async0();
      }
      __syncthreads();
#pragma unroll
      for (int kk = 0; kk < GEMM_KC; kk += 4) {
        v2f a, b;
        a.x = As[buf][wm * 16 + row][kk + kp];
        a.y = As[buf][wm * 16 + row][kk + kp + 1];
        b.x = Ws[buf][wn * 16 + row][kk + kp];
        b.y = Ws[buf][wn * 16 + row][kk + kp + 1];
        acc = __builtin_amdgcn_wmma_f32_16x16x4_f32(false, a, false, b,
                                                    (short)0, acc, false, false);
      }
      __syncthreads();
      buf ^= 1;
    }
  } else {
    for (int k0 = 0; k0 < K; k0 += GEMM_KC) {
      for (int i = tid; i < 32 * GEMM_KC; i += 256) {
        int r = i / GEMM_KC, c = i % GEMM_KC;
        int m = m0 + r, k = k0 + c;
        As[0][r][c] = (m < M && k < K) ? X[(size_t)m * K + k] : 0.0f;
      }
      for (int i = tid; i < 64 * GEMM_KC; i += 256) {
        int r = i / GEMM_KC, c = i % GEMM_KC;
        int n = n0 + r, k = k0 + c;
        Ws[0][r][c] = (n < N && k < K) ? W[(size_t)n * K + k] : 0.0f;
      }
      __syncthreads();
#pragma unroll
      for (int kk = 0; kk < GEMM_KC; kk += 4) {
        v2f a, b;
        a.x = As[0][wm * 16 + row][kk + kp];
        a.y = As[0][wm * 16 + row][kk + kp + 1];
        b.x = Ws[0][wn * 16 + row][kk + kp];
        b.y = Ws[0][wn * 16 + row][kk + kp + 1];
        acc = __builtin_amdgcn_wmma_f32_16x16x4_f32(false, a, false, b,
                                                    (short)0, acc, false, false);
      }
      __syncthreads();
    }
  }

  const int n  = n0 + wn * 16 + (lane & 15);
  const int mb = m0 + wm * 16 + ((lane >> 4) << 3);
  if (n < N) {
    const float bv = bias[n];
#pragma unroll
    for (int r = 0; r < 8; ++r) {
      int m = mb + r;
      if (m < M) {
        float v = acc[r] + bv;
        if (relu) v = fmaxf(v, 0.0f);
        Y[(size_t)m * N + n] = v;
      }
    }
  }
}

// ---------------------------------------------------------------------------
// Small utility: zero a float region (BN statistics accumulators).
// ---------------------------------------------------------------------------
__global__ void zero_kernel(float* __restrict__ p, int n) {
  int i = blockIdx.x * blockDim.x + threadIdx.x;
  if (i < n) p[i] = 0.0f;
}

// ---------------------------------------------------------------------------
// Wave+block reduction of per-thread (sum, sumsq) vectors into global stats.
// stats layout: stats[c] = sum, stats[C+c] = sumsq.
// ---------------------------------------------------------------------------
template <int C>
__device__ __forceinline__ void accum_stats(float* __restrict__ stats,
                                            float (&s)[C], float (&q)[C],
                                            float* racc, int tid) {
#pragma unroll
  for (int c = 0; c < C; ++c) {
    for (int o = 16; o > 0; o >>= 1) {
      s[c] += __shfl_down(s[c], o, 32);
      q[c] += __shfl_down(q[c], o, 32);
    }
  }
  if ((tid & 31) == 0) {
#pragma unroll
    for (int c = 0; c < C; ++c) {
      atomicAdd(&racc[c], s[c]);
      atomicAdd(&racc[C + c], q[c]);
    }
  }
  __syncthreads();
  if (tid < 2 * C) atomicAdd(&stats[tid], racc[tid]);
}

// ---------------------------------------------------------------------------
// conv1: [B,1,155,220] (*) [4,1,11,11] -> relu -> [B,4,145,210] + BN stats.
// 16x16 output tile per block; 26x26 input tile staged via async-to-LDS.
// ---------------------------------------------------------------------------
__global__ __launch_bounds__(256)
void conv1_kernel(const float* __restrict__ in, const float* __restrict__ w,
                  const float* __restrict__ bias, float* __restrict__ y,
                  float* __restrict__ stats) {
  const int IH = 155, IW = 220, OH = 145, OW = 210;
  __shared__ float tin[26][27];
  __shared__ float tw[4][121];
  __shared__ float sb[4];
  __shared__ float racc[8];

  const int tid = threadIdx.x;
  const int b   = blockIdx.z;
  const int ox0 = blockIdx.x * 16, oy0 = blockIdx.y * 16;

  for (int i = tid; i < 484; i += 256) tw[i / 121][i % 121] = w[i];
  if (tid < 4) sb[tid] = bias[tid];
  if (tid < 8) racc[tid] = 0.0f;

  const float* gin = in + (size_t)b * IH * IW;
  for (int i = tid; i < 26 * 26; i += 256) {
    int r = i / 26, c = i % 26;
    int iy = min(oy0 + r, IH - 1);
    int ix = min(ox0 + c, IW - 1);
    async_ld_f32(&tin[r][c], gin + iy * IW + ix);
  }
  wait_async0();
  __syncthreads();

  const int ly = tid >> 4, lx = tid & 15;
  const int oy = oy0 + ly, ox = ox0 + lx;
  const bool valid = (oy < OH) && (ox < OW);

  float acc[4] = {sb[0], sb[1], sb[2], sb[3]};
#pragma unroll
  for (int ky = 0; ky < 11; ++ky)
#pragma unroll
    for (int kx = 0; kx < 11; ++kx) {
      float v = tin[ly + ky][lx + kx];
#pragma unroll
      for (int c = 0; c < 4; ++c) acc[c] = fmaf(v, tw[c][ky * 11 + kx], acc[c]);
    }

  float s[4], q[4];
#pragma unroll
  for (int c = 0; c < 4; ++c) {
    float v = fmaxf(acc[c], 0.0f);
    if (valid) y[(((size_t)b * 4 + c) * OH + oy) * OW + ox] = v;
    s[c] = valid ? v : 0.0f;
    q[c] = valid ? v * v : 0.0f;
  }
  accum_stats<4>(stats, s, q, racc, tid);
}

// ---------------------------------------------------------------------------
// conv2: [B,4,72,105] (*) [4,4,5,5] -> relu -> [B,4,68,101] + BN stats.
// ---------------------------------------------------------------------------
__global__ __launch_bounds__(256)
void conv2_kernel(const float* __restrict__ in, const float* __restrict__ w,
                  const float* __restrict__ bias, float* __restrict__ y,
                  float* __restrict__ stats) {
  const int IH = 72, IW = 105, OH = 68, OW = 101;
  __shared__ float tin[4][20][21];
  __shared__ float tw[4][100];     // [oc][ic*25 + ky*5 + kx]
  __shared__ float sb[4];
  __shared__ float racc[8];

  const int tid = threadIdx.x;
  const int b   = blockIdx.z;
  const int ox0 = blockIdx.x * 16, oy0 = blockIdx.y * 16;

  for (int i = tid; i < 400; i += 256) tw[i / 100][i % 100] = w[i];
  if (tid < 4) sb[tid] = bias[tid];
  if (tid < 8) racc[tid] = 0.0f;

  const float* gin = in + (size_t)b * 4 * IH * IW;
  for (int i = tid; i < 4 * 20 * 20; i += 256) {
    int ic = i / 400, rem = i % 400;
    int r = rem / 20, c = rem % 20;
    int iy = min(oy0 + r, IH - 1);
    int ix = min(ox0 + c, IW - 1);
    async_ld_f32(&tin[ic][r][c], gin + ((size_t)ic * IH + iy) * IW + ix);
  }
  wait_async0();
  __syncthreads();

  const int ly = tid >> 4, lx = tid & 15;
  const int oy = oy0 + ly, ox = ox0 + lx;
  const bool valid = (oy < OH) && (ox < OW);

  float acc[4] = {sb[0], sb[1], sb[2], sb[3]};
#pragma unroll
  for (int ic = 0; ic < 4; ++ic)
#pragma unroll
    for (int ky = 0; ky < 5; ++ky)
#pragma unroll
      for (int kx = 0; kx < 5; ++kx) {
        float v = tin[ic][ly + ky][lx + kx];
#pragma unroll
        for (int c = 0; c < 4; ++c)
          acc[c] = fmaf(v, tw[c][ic * 25 + ky * 5 + kx], acc[c]);
      }

  float s[4], q[4];
#pragma unroll
  for (int c = 0; c < 4; ++c) {
    float v = fmaxf(acc[c], 0.0f);
    if (valid) y[(((size_t)b * 4 + c) * OH + oy) * OW + ox] = v;
    s[c] = valid ? v : 0.0f;
    q[c] = valid ? v * v : 0.0f;
  }
  accum_stats<4>(stats, s, q, racc, tid);
}

// ---------------------------------------------------------------------------
// conv3: [B,4,34,50] (*) [8,4,3,3] -> relu -> [B,8,32,48] + BN stats.
// ---------------------------------------------------------------------------
__global__ __launch_bounds__(256)
void conv3_kernel(const float* __restrict__ in, const float* __restrict__ w,
                  const float* __restrict__ bias, float* __restrict__ y,
                  float* __restrict__ stats) {
  const int IH = 34, IW = 50, OH = 32, OW = 48;
  __shared__ float tin[4][18][19];
  __shared__ float tw[8][36];     // [oc][ic*9 + ky*3 + kx]
  __shared__ float sb[8];
  __shared__ float racc[16];

  const int tid = threadIdx.x;
  const int b   = blockIdx.z;
  const int ox0 = blockIdx.x * 16, oy0 = blockIdx.y * 16;

  for (int i = tid; i < 288; i += 256) tw[i / 36][i % 36] = w[i];
  if (tid < 8) sb[tid] = bias[tid];
  if (tid < 16) racc[tid] = 0.0f;

  const float* gin = in + (size_t)b * 4 * IH * IW;
  for (int i = tid; i < 4 * 18 * 18; i += 256) {
    int ic = i / 324, rem = i % 324;
    int r = rem / 18, c = rem % 18;
    int iy = min(oy0 + r, IH - 1);
    int ix = min(ox0 + c, IW - 1);
    async_ld_f32(&tin[ic][r][c], gin + ((size_t)ic * IH + iy) * IW + ix);
  }
  wait_async0();
  __syncthreads();

  const int ly = tid >> 4, lx = tid & 15;
  const int oy = oy0 + ly, ox = ox0 + lx;
  const bool valid = (oy < OH) && (ox < OW);

  float acc[8] = {sb[0], sb[1], sb[2], sb[3], sb[4], sb[5], sb[6], sb[7]};
#pragma unroll
  for (int ic = 0; ic < 4; ++ic)
#pragma unroll
    for (int ky = 0; ky < 3; ++ky)
#pragma unroll
      for (int kx = 0; kx < 3; ++kx) {
        float v = tin[ic][ly + ky][lx + kx];
#pragma unroll
        for (int c = 0; c < 8; ++c)
          acc[c] = fmaf(v, tw[c][ic * 9 + ky * 3 + kx], acc[c]);
      }

  float s[8], q[8];
#pragma unroll
  for (int c = 0; c < 8; ++c) {
    float v = fmaxf(acc[c], 0.0f);
    if (valid) y[(((size_t)b * 8 + c) * OH + oy) * OW + ox] = v;
    s[c] = valid ? v : 0.0f;
    q[c] = valid ? v * v : 0.0f;
  }
  accum_stats<8>(stats, s, q, racc, tid);
}

// ---------------------------------------------------------------------------
// Fused train-mode BatchNorm (batch stats) + 2x2/2 max pool.
// in: [B,C,H,W] -> out: [B,C,Ho,Wo], Ho=H/2, Wo=W/2.
// ---------------------------------------------------------------------------
__global__ __launch_bounds__(256)
void bnpool_kernel(const float* __restrict__ yin, const float* __restrict__ stats,
                   const float* __restrict__ g, const float* __restrict__ be,
                   float* __restrict__ out,
                   int B, int C, int H, int W, int Ho, int Wo) {
  int idx = blockIdx.x * 256 + threadIdx.x;
  int total = B * C * Ho * Wo;
  if (idx >= total) return;
  int wo = idx % Wo;
  int ho = (idx / Wo) % Ho;
  int c  = (idx / (Wo * Ho)) % C;
  int b  = idx / (Wo * Ho * C);

  float cnt  = (float)B * (float)H * (float)W;
  float mean = stats[c] / cnt;
  float var  = stats[C + c] / cnt - mean * mean;
  float sc   = g[c] * rsqrtf(var + 1e-5f);
  float sh   = be[c] - mean * sc;

  size_t base = (((size_t)b * C + c) * H + 2 * ho) * W + 2 * wo;
  float v0 = fmaf(yin[base], sc, sh);
  float v1 = fmaf(yin[base + 1], sc, sh);
  float v2 = fmaf(yin[base + W], sc, sh);
  float v3 = fmaf(yin[base + W + 1], sc, sh);
  out[idx] = fmaxf(fmaxf(v0, v1), fmaxf(v2, v3));
}

// ---------------------------------------------------------------------------
// Batched 8-qubit real state-vector sim. One block (256 threads) per batch
// element; state lives in LDS, one amplitude per thread. Qubit q's bit is
// bit (7-q) of the amplitude index (reference uses MSB-first qubits).
// ---------------------------------------------------------------------------
__device__ __forceinline__ void gate1(float* st, int i, int q,
                                      float m00, float m01, float m10, float m11) {
  int mask = 1 << (7 - q);
  float a = st[i];
  float p = st[i ^ mask];
  __syncthreads();
  st[i] = ((i & mask) == 0) ? fmaf(m00, a, m01 * p) : fmaf(m11, a, m10 * p);
  __syncthreads();
}

__device__ __forceinline__ void cnot_gate(float* st, int i, int cq, int tq) {
  int cm = 1 << (7 - cq), tm = 1 << (7 - tq);
  float a = st[i];
  float p = st[i ^ tm];
  __syncthreads();
  st[i] = (i & cm) ? p : a;
  __syncthreads();
}

__global__ __launch_bounds__(256)
void qsim_kernel(const float* __restrict__ x1, const float* __restrict__ x2,
                 const float* __restrict__ qp,
                 float* __restrict__ z1, float* __restrict__ z2) {
  __shared__ float st[256];
  __shared__ float red[256];
  const int i = threadIdx.x, b = blockIdx.x;

  st[i] = (i == 0) ? 1.0f : 0.0f;
  __syncthreads();

  // RY data encoding: x1 on qubits 0..3, x2 on qubits 4..7
  for (int qq = 0; qq < 4; ++qq) {
    float t = 0.5f * x1[b * 4 + qq];
    float c = cosf(t), s = sinf(t);
    gate1(st, i, qq, c, -s, s, c);
    t = 0.5f * x2[b * 4 + qq];
    c = cosf(t); s = sinf(t);
    gate1(st, i, qq + 4, c, -s, s, c);
  }
  // shared trainable RY on both channels
  for (int qq = 0; qq < 4; ++qq) {
    float t = 0.5f * qp[qq];
    float c = cosf(t), s = sinf(t);
    gate1(st, i, qq, c, -s, s, c);
    gate1(st, i, qq + 4, c, -s, s, c);
  }
  // CNOT interference
  for (int qq = 0; qq < 4; ++qq) cnot_gate(st, i, qq, qq + 4);
  // final Hadamards
  const float k = 0.70710678118654752440f;
  for (int qq = 0; qq < 8; ++qq) gate1(st, i, qq, k, k, k, -k);

  // <Z_q> = sum_i (+-) |amp_i|^2
  float p2 = st[i] * st[i];
  for (int qq = 0; qq < 8; ++qq) {
    red[i] = ((i >> (7 - qq)) & 1) ? -p2 : p2;
    __syncthreads();
    for (int o = 128; o > 0; o >>= 1) {
      if (i < o) red[i] += red[i + o];
      __syncthreads();
    }
    if (i == 0) {
      if (qq < 4) z1[b * 4 + qq] = red[0];
      else        z2[b * 4 + (qq - 4)] = red[0];
    }
    __syncthreads();
  }
}

// ---------------------------------------------------------------------------
// Host-side orchestration.
// ---------------------------------------------------------------------------
extern "C" void kernel_launch(void* const* d_in, const int* in_sizes, int n_in,
                              void* d_out, int out_size, void* d_ws, size_t ws_size,
                              hipStream_t stream) {
  (void)in_sizes; (void)n_in; (void)out_size; (void)ws_size;

  const float* input[2] = {(const float*)d_in[0], (const float*)d_in[1]};
  const float* cw1 = (const float*)d_in[2];  const float* cb1 = (const float*)d_in[3];
  const float* g1  = (const float*)d_in[4];  const float* be1 = (const float*)d_in[5];
  const float* cw2 = (const float*)d_in[6];  const float* cb2 = (const float*)d_in[7];
  const float* g2  = (const float*)d_in[8];  const float* be2 = (const float*)d_in[9];
  const float* cw3 = (const float*)d_in[10]; const float* cb3 = (const float*)d_in[11];
  const float* g3  = (const float*)d_in[12]; const float* be3 = (const float*)d_in[13];
  const float* fw1 = (const float*)d_in[14]; const float* fb1 = (const float*)d_in[15];
  const float* fw2 = (const float*)d_in[16]; const float* fb2 = (const float*)d_in[17];
  const float* fw3 = (const float*)d_in[18]; const float* fb3 = (const float*)d_in[19];
  const float* pw1 = (const float*)d_in[20]; const float* pb1 = (const float*)d_in[21];
  const float* pw2 = (const float*)d_in[22]; const float* pb2 = (const float*)d_in[23];
  const float* pw3 = (const float*)d_in[24]; const float* pb3 = (const float*)d_in[25];
  const float* qp  = (const float*)d_in[26];

  float* ws = (float*)d_ws;
  // Workspace layout (floats). Branches run sequentially; R0/R1 ping-pong.
  float* R0   = ws;                       // max 256*4*145*210 = 31,180,800
  float* R1   = R0 + 31180800;            // max 256*4*72*105  =  7,741,440
  float* FC1  = R1 + 7741440;             // 256*1024
  float* FC2  = FC1 + 262144;             // 256*128
  float* O1   = FC2 + 32768;              // 256*4
  float* O2   = O1 + 1024;                // 256*4
  float* Q1   = O2 + 1024;                // 256*4 (padded)
  float* Q2   = Q1 + 1024;
  float* P1   = Q2 + 1024;                // 256*16
  float* P2   = P1 + 4096;                // 256*32
  float* STAT = P2 + 8192;                // 32 floats: st1[8], st2[8], st3[16]

  const int B = 256;

  for (int br = 0; br < 2; ++br) {
    float* obuf = (br == 0) ? O1 : O2;

    zero_kernel<<<1, 32, 0, stream>>>(STAT, 32);

    conv1_kernel<<<dim3(14, 10, B), 256, 0, stream>>>(input[br], cw1, cb1, R0, STAT);
    {
      int total = B * 4 * 72 * 105;
      bnpool_kernel<<<(total + 255) / 256, 256, 0, stream>>>(
          R0, STAT, g1, be1, R1, B, 4, 145, 210, 72, 105);
    }
    conv2_kernel<<<dim3(7, 5, B), 256, 0, stream>>>(R1, cw2, cb2, R0, STAT + 8);
    {
      int total = B * 4 * 34 * 50;
      bnpool_kernel<<<(total + 255) / 256, 256, 0, stream>>>(
          R0, STAT + 8, g2, be2, R1, B, 4, 68, 101, 34, 50);
    }
    conv3_kernel<<<dim3(3, 2, B), 256, 0, stream>>>(R1, cw3, cb3, R0, STAT + 16);
    {
      int total = B * 8 * 16 * 24;
      bnpool_kernel<<<(total + 255) / 256, 256, 0, stream>>>(
          R0, STAT + 16, g3, be3, R1, B, 8, 32, 48, 16, 24);
    }

    // MLP head (WMMA f32): 3072 -> 1024 -> 128 -> 4
    wmma_gemm_bias_act<<<dim3(8, 16), 256, 0, stream>>>(R1, fw1, fb1, FC1, B, 1024, 3072, 1);
    wmma_gemm_bias_act<<<dim3(8, 2),  256, 0, stream>>>(FC1, fw2, fb2, FC2, B, 128, 1024, 1);
    wmma_gemm_bias_act<<<dim3(8, 1),  256, 0, stream>>>(FC2, fw3, fb3, obuf, B, 4, 128, 0);
  }

  // Quantum interference block: one block per batch element.
  qsim_kernel<<<B, 256, 0, stream>>>(O1, O2, qp, Q1, Q2);

  // Shared post-MLP 4 -> 16 -> 32 -> 64 on q1 and q2 (WMMA f32).
  float* out = (float*)d_out;
  for (int h = 0; h < 2; ++h) {
    const float* qin = (h == 0) ? Q1 : Q2;
    float* dst = out + (size_t)h * B * 64;
    wmma_gemm_bias_act<<<dim3(8, 1), 256, 0, stream>>>(qin, pw1, pb1, P1, B, 16, 4, 1);
    wmma_gemm_bias_act<<<dim3(8, 1), 256, 0, stream>>>(P1, pw2, pb2, P2, B, 32, 16, 1);
    wmma_gemm_bias_act<<<dim3(8, 1), 256, 0, stream>>>(P2, pw3, pb3, dst, B, 64, 32, 0);
  }
}